// EmbGraph_83107617178467
// MI455X (gfx1250) — compile-verified
//
#include <hip/hip_runtime.h>

typedef float v2f __attribute__((ext_vector_type(2)));
typedef float v8f __attribute__((ext_vector_type(8)));

#define GCN_D 128
#define ROWS_PER_BLK 64   // 4 row-tiles of 16 per block

// ---------------------------------------------------------------------------
// Degree / normalization kernels
// ---------------------------------------------------------------------------
__global__ void deg_init_kernel(float* deg, int n) {
    int i = blockIdx.x * blockDim.x + threadIdx.x;
    if (i < n) deg[i] = 1.0f;                       // self-loop contributes 1
}

__global__ void deg_edge_kernel(const int* __restrict__ dst, float* deg, int e) {
    int i = blockIdx.x * blockDim.x + threadIdx.x;
    if (i < e) atomicAdd(&deg[dst[i]], 1.0f);
}

__global__ void deg_rsqrt_kernel(float* deg, int n) {
    int i = blockIdx.x * blockDim.x + threadIdx.x;
    if (i < n) deg[i] = rsqrtf(deg[i]);             // deg -> dinv in place
}

// ---------------------------------------------------------------------------
// WMMA GEMM: C[N x 128] = act(A)[N x 128] * W[128 x 128]
// 256 threads = 8 waves per block; block covers 64 rows x 128 cols.
// Each wave owns one 16-col stripe and 4 row-tiles (4 x 16x16 accumulators),
// so each B fragment feeds 4 independent WMMAs (breaks accumulator RAW chain
// and amortizes LDS traffic). W (64 KB) + A tile (32 KB) staged in LDS.
// fp32 WMMA (V_WMMA_F32_16X16X4_F32) keeps full reference precision.
// ---------------------------------------------------------------------------
__global__ __launch_bounds__(256)
void gcn_gemm_wmma(const float* __restrict__ A, const float* __restrict__ W,
                   float* __restrict__ C, int n, int applyRelu)
{
    __shared__ float sA[ROWS_PER_BLK * GCN_D];   // 32 KB
    __shared__ float sW[GCN_D * GCN_D];          // 64 KB

    const int tid = threadIdx.x;
    const int rowBase = blockIdx.x * ROWS_PER_BLK;

    // Stage A tile (64x128) with optional ReLU; 2048 float4 loads
    {
        float4* dst4 = (float4*)sA;
        for (int i = tid; i < (ROWS_PER_BLK * GCN_D) / 4; i += 256) {
            int r = i / (GCN_D / 4);
            int c = i % (GCN_D / 4);
            int row = rowBase + r;
            if (row >= n) row = n - 1;    // clamp for the tail block
            float4 v = ((const float4*)(A + (size_t)row * GCN_D))[c];
            if (applyRelu) {
                v.x = fmaxf(v.x, 0.f); v.y = fmaxf(v.y, 0.f);
                v.z = fmaxf(v.z, 0.f); v.w = fmaxf(v.w, 0.f);
            }
            dst4[i] = v;
        }
    }
    // Stage W (128x128): 4096 float4 loads
    {
        const float4* src4 = (const float4*)W;
        float4* dst4 = (float4*)sW;
        for (int i = tid; i < (GCN_D * GCN_D) / 4; i += 256) dst4[i] = src4[i];
    }
    __syncthreads();

    const int lane = tid & 31;
    const int wave = tid >> 5;
    const int m  = lane & 15;             // M index (A) / N index (B)
    const int hi = lane >> 4;             // half-wave selector
    const int colBase = wave * 16;

    v8f acc0 = {}, acc1 = {}, acc2 = {}, acc3 = {};

    #pragma unroll
    for (int k0 = 0; k0 < GCN_D; k0 += 4) {
        const int k = k0 + 2 * hi;
        // B fragment: VGPR v <- W[k0 + v + 2*hi][colBase + m]  (shared by 4 WMMAs)
        v2f b;
        b.x = sW[k * GCN_D + colBase + m];
        b.y = sW[(k + 1) * GCN_D + colBase + m];
        // A fragments for the 4 row-tiles: VGPR v <- A[rt*16 + m][k0 + v + 2*hi]
        v2f a0, a1, a2, a3;
        a0.x = sA[(0 * 16 + m) * GCN_D + k]; a0.y = sA[(0 * 16 + m) * GCN_D + k + 1];
        a1.x = sA[(1 * 16 + m) * GCN_D + k]; a1.y = sA[(1 * 16 + m) * GCN_D + k + 1];
        a2.x = sA[(2 * 16 + m) * GCN_D + k]; a2.y = sA[(2 * 16 + m) * GCN_D + k + 1];
        a3.x = sA[(3 * 16 + m) * GCN_D + k]; a3.y = sA[(3 * 16 + m) * GCN_D + k + 1];

        acc0 = __builtin_amdgcn_wmma_f32_16x16x4_f32(false, a0, false, b, (short)0, acc0, false, false);
        acc1 = __builtin_amdgcn_wmma_f32_16x16x4_f32(false, a1, false, b, (short)0, acc1, false, false);
        acc2 = __builtin_amdgcn_wmma_f32_16x16x4_f32(false, a2, false, b, (short)0, acc2, false, false);
        acc3 = __builtin_amdgcn_wmma_f32_16x16x4_f32(false, a3, false, b, (short)0, acc3, false, false);
    }

    // C/D layout: VGPR v holds row (rt*16 + v + 8*hi), col m of the tile
    #pragma unroll
    for (int v = 0; v < 8; ++v) {
        int r0 = rowBase + 0 * 16 + v + 8 * hi;
        int r1 = rowBase + 1 * 16 + v + 8 * hi;
        int r2 = rowBase + 2 * 16 + v + 8 * hi;
        int r3 = rowBase + 3 * 16 + v + 8 * hi;
        if (r0 < n) C[(size_t)r0 * GCN_D + colBase + m] = acc0[v];
        if (r1 < n) C[(size_t)r1 * GCN_D + colBase + m] = acc1[v];
        if (r2 < n) C[(size_t)r2 * GCN_D + colBase + m] = acc2[v];
        if (r3 < n) C[(size_t)r3 * GCN_D + colBase + m] = acc3[v];
    }
}

// ---------------------------------------------------------------------------
// Aggregation init: folds self-loop (norm = dinv[i]^2) and bias.
// agg[i][d] = hW[i][d] * dinv[i]^2 + b[d]     (full overwrite -> deterministic)
// ---------------------------------------------------------------------------
__global__ __launch_bounds__(256)
void agg_init_kernel(const float* __restrict__ hW, const float* __restrict__ dinv,
                     const float* __restrict__ bias, float* __restrict__ agg, int n)
{
    int i4 = blockIdx.x * blockDim.x + threadIdx.x;       // float4 index
    int total4 = n * (GCN_D / 4);
    if (i4 >= total4) return;
    int node = i4 >> 5;                                   // /(128/4)
    int d4   = i4 & 31;
    float di = dinv[node];
    float w  = di * di;
    float4 v = ((const float4*)hW)[i4];
    float4 b = ((const float4*)bias)[d4];
    float4 o;
    o.x = v.x * w + b.x; o.y = v.y * w + b.y;
    o.z = v.z * w + b.z; o.w = v.w * w + b.w;
    ((float4*)agg)[i4] = o;
}

// ---------------------------------------------------------------------------
// Edge scatter: one wave per edge, float4 per lane (32 lanes * 4 = 128 feats)
// agg[dst] += hW[src] * dinv[src]*dinv[dst]
// ---------------------------------------------------------------------------
__global__ __launch_bounds__(256)
void scatter_edge_kernel(const float* __restrict__ hW, const float* __restrict__ dinv,
                         const int* __restrict__ src, const int* __restrict__ dst,
                         float* __restrict__ agg, int e)
{
    int edge = blockIdx.x * 8 + (threadIdx.x >> 5);
    if (edge >= e) return;
    int lane = threadIdx.x & 31;
    int s = src[edge];
    int d = dst[edge];
    float w = dinv[s] * dinv[d];
    float4 v = ((const float4*)(hW + (size_t)s * GCN_D))[lane];
    float* out = agg + (size_t)d * GCN_D + lane * 4;
    atomicAdd(out + 0, v.x * w);
    atomicAdd(out + 1, v.y * w);
    atomicAdd(out + 2, v.z * w);
    atomicAdd(out + 3, v.w * w);
}

// ---------------------------------------------------------------------------
// Residual: out += x
// ---------------------------------------------------------------------------
__global__ __launch_bounds__(256)
void residual_add_kernel(const float* __restrict__ x, float* __restrict__ out, int total4)
{
    int i4 = blockIdx.x * blockDim.x + threadIdx.x;
    if (i4 >= total4) return;
    float4 a = ((const float4*)x)[i4];
    float4 o = ((float4*)out)[i4];
    o.x += a.x; o.y += a.y; o.z += a.z; o.w += a.w;
    ((float4*)out)[i4] = o;
}

// ---------------------------------------------------------------------------
extern "C" void kernel_launch(void* const* d_in, const int* in_sizes, int n_in,
                              void* d_out, int out_size, void* d_ws, size_t ws_size,
                              hipStream_t stream)
{
    const float* x  = (const float*)d_in[0];
    const int*   ei = (const int*)d_in[1];
    const float* W0 = (const float*)d_in[2];
    const float* b0 = (const float*)d_in[3];
    const float* W1 = (const float*)d_in[4];
    const float* b1 = (const float*)d_in[5];
    const float* W2 = (const float*)d_in[6];
    const float* b2 = (const float*)d_in[7];

    const int N = in_sizes[0] / GCN_D;
    const int E = in_sizes[1] / 2;
    const int* srcIdx = ei;           // edge_index[0]
    const int* dstIdx = ei + E;       // edge_index[1]

    float* out  = (float*)d_out;
    float* bufA = (float*)d_ws;                       // hW scratch
    float* bufB = bufA + (size_t)N * GCN_D;           // agg / next-layer input
    float* dinv = bufB + (size_t)N * GCN_D;           // N floats (deg -> dinv)

    // Degree + symmetric normalization
    deg_init_kernel <<<(N + 255) / 256, 256, 0, stream>>>(dinv, N);
    deg_edge_kernel <<<(E + 255) / 256, 256, 0, stream>>>(dstIdx, dinv, E);
    deg_rsqrt_kernel<<<(N + 255) / 256, 256, 0, stream>>>(dinv, N);

    const float* Ws[3] = {W0, W1, W2};
    const float* bs[3] = {b0, b1, b2};

    const int total4   = N * (GCN_D / 4);
    const int gemmGrid = (N + ROWS_PER_BLK - 1) / ROWS_PER_BLK;

    const float* hin = x;
    for (int l = 0; l < 3; ++l) {
        float* hw  = bufA;
        float* agg = (l == 2) ? out : bufB;
        // GEMM (ReLU applied to input of layers 1 and 2)
        gcn_gemm_wmma<<<gemmGrid, 256, 0, stream>>>(hin, Ws[l], hw, N, l > 0 ? 1 : 0);
        // agg = hW * dinv^2 + b   (self-loop + bias, full overwrite)
        agg_init_kernel<<<(total4 + 255) / 256, 256, 0, stream>>>(hw, dinv, bs[l], agg, N);
        // agg[dst] += hW[src] * norm
        scatter_edge_kernel<<<(E + 7) / 8, 256, 0, stream>>>(hw, dinv, srcIdx, dstIdx, agg, E);
        hin = agg;
    }

    // out = agg2 + x
    residual_add_kernel<<<(total4 + 255) / 256, 256, 0, stream>>>(x, out, total4);
}